// CopyGenerator_17463337025745
// MI455X (gfx1250) — compile-verified
//
#include <hip/hip_runtime.h>
#include <hip/hip_bf16.h>
#include <math.h>

// Problem constants (from the reference)
#define TLEN 64
#define BSZ  32
#define HID  1024
#define SLEN 200
#define VT   32000
#define VE   2000
#define NROW (TLEN * BSZ)      // 2048
#define VOUT (VT + VE)         // 34000
#define NPART 400              // 200 col-wgs * 2 N-waves of per-row softmax partials

typedef __attribute__((ext_vector_type(2))) float v2f;
typedef __attribute__((ext_vector_type(8))) float v8f;

// ---------------------------------------------------------------------------
// Kernel 1: copy gate.  One wave per row: copy = sigmoid(h.w_copy + b_copy).
// Emits copy, safe_log(copy), and (1-copy).
// ---------------------------------------------------------------------------
__global__ void gate_kernel(const float* __restrict__ h,
                            const float* __restrict__ w_copy,
                            const float* __restrict__ b_copy,
                            float* __restrict__ copyv,
                            float* __restrict__ logcopy,
                            float* __restrict__ oneminus) {
    int wave = (blockIdx.x * blockDim.x + threadIdx.x) >> 5;   // row id, 0..2047
    int lane = threadIdx.x & 31;
    const float* hr = h + (size_t)wave * HID;
    float s = 0.f;
    #pragma unroll 4
    for (int i = lane; i < HID; i += 32) s += hr[i] * w_copy[i];
    #pragma unroll
    for (int m = 16; m >= 1; m >>= 1) s += __shfl_xor(s, m, 32);
    if (lane == 0) {
        float z = s + b_copy[0];
        float c = 1.f / (1.f + __expf(-z));
        copyv[wave]    = c;
        float cc       = fminf(fmaxf(c, 0.001f), 0.999f);
        logcopy[wave]  = __logf(cc);
        oneminus[wave] = 1.f - c;
    }
}

// ---------------------------------------------------------------------------
// Single WMMA GEMM pass: logits = h[2048,1024] @ W_out[32000,1024]^T + b_out.
//  * wave tile 64x80: 4 M-tiles x 5 N-tiles of V_WMMA_F32_16X16X4_F32
//    (20 x v8f accumulators; 9 b64 loads per 20 WMMAs)
//  * workgroup = 8 waves arranged 4(M) x 2(N) -> 256x160 tile
//    grid (200, 8): 32000 = 200*160, 2048 = 8*256 (exact, EXEC all-1s)
//  * writes RAW logits into the V_TGT region of d_out, and per-row
//    (max, sum-exp) partials for the log_softmax correction.
// ---------------------------------------------------------------------------
__global__ __launch_bounds__(256) void gemm_kernel(
        const float* __restrict__ h,
        const float* __restrict__ W,
        const float* __restrict__ b_out,
        float* __restrict__ out,
        float* __restrict__ pM,
        float* __restrict__ pS) {
    const int wv   = threadIdx.x >> 5;     // wave in block, 0..7
    const int lane = threadIdx.x & 31;
    const int lo   = lane & 15;            // position within 16-lane half
    const int hi   = lane >> 4;            // half-wave select
    const int msub = wv >> 1;              // 0..3  (M sub-block of 64 rows)
    const int nsub = wv & 1;               // 0..1  (N sub-block of 80 cols)
    const int m0   = blockIdx.y * 256 + msub * 64;
    const int v0   = blockIdx.x * 160 + nsub * 80;

    // A fragment source (ISA 16x4 f32 layout): h[m0+16*mt+lo][k + 2*hi + {0,1}]
    const float* aptr = h + (size_t)(m0 + lo) * HID + 2 * hi;
    // B fragment source (mirror layout): W[v0+16*i+lo][k + 2*hi + {0,1}]
    const float* bptr = W + (size_t)(v0 + lo) * HID + 2 * hi;

    v8f acc[4][5] = {};

    for (int k = 0; k < HID; k += 4) {
        v2f a[4], b[5];
        #pragma unroll
        for (int mt = 0; mt < 4; ++mt)
            a[mt] = *(const v2f*)(aptr + (size_t)mt * 16 * HID + k);
        #pragma unroll
        for (int i = 0; i < 5; ++i)
            b[i] = *(const v2f*)(bptr + (size_t)i * 16 * HID + k);
        #pragma unroll
        for (int mt = 0; mt < 4; ++mt)
            #pragma unroll
            for (int i = 0; i < 5; ++i)
                acc[mt][i] = __builtin_amdgcn_wmma_f32_16x16x4_f32(
                    false, a[mt], false, b[i], (short)0, acc[mt][i], false, false);
    }

    // Bias per column tile (same for every C vgpr of the tile).
    float bb[5];
    #pragma unroll
    for (int i = 0; i < 5; ++i) bb[i] = b_out[v0 + 16 * i + lo];

    const int pidx = blockIdx.x * 2 + nsub;          // 0..399

    #pragma unroll
    for (int mt = 0; mt < 4; ++mt) {
        #pragma unroll
        for (int r = 0; r < 8; ++r) {
            int row = m0 + mt * 16 + r + 8 * hi;     // C/D layout: vgpr r, halves 0/8
            size_t base = (size_t)row * VOUT + v0 + lo;
            float v[5];
            float rmax = -1e30f;
            #pragma unroll
            for (int i = 0; i < 5; ++i) {
                v[i] = acc[mt][i][r] + bb[i];
                rmax = fmaxf(rmax, v[i]);
                out[base + 16 * i] = v[i];           // raw logit (fixed up later)
            }
            #pragma unroll
            for (int m = 8; m >= 1; m >>= 1)
                rmax = fmaxf(rmax, __shfl_xor(rmax, m, 16));
            float rsum = 0.f;
            #pragma unroll
            for (int i = 0; i < 5; ++i) rsum += __expf(v[i] - rmax);
            #pragma unroll
            for (int m = 8; m >= 1; m >>= 1)
                rsum += __shfl_xor(rsum, m, 16);
            if (lo == 0) {
                pM[(size_t)row * NPART + pidx] = rmax;
                pS[(size_t)row * NPART + pidx] = rsum;
            }
        }
    }
}

// ---------------------------------------------------------------------------
// Kernel 3: fold 400 (max, sumexp) partials per row into the correction term
// corr[row] = safe_log(copy) - rowMax - log(rowSumExp).
// ---------------------------------------------------------------------------
__global__ void stats_reduce_kernel(const float* __restrict__ pM,
                                    const float* __restrict__ pS,
                                    const float* __restrict__ logcopy,
                                    float* __restrict__ corr) {
    int n = blockIdx.x * blockDim.x + threadIdx.x;
    if (n >= NROW) return;
    const float* m = pM + (size_t)n * NPART;
    const float* s = pS + (size_t)n * NPART;
    float M = -1e30f;
    for (int j = 0; j < NPART; ++j) M = fmaxf(M, m[j]);
    float S = 0.f;
    for (int j = 0; j < NPART; ++j) S += s[j] * __expf(m[j] - M);
    corr[n] = logcopy[n] - M - __logf(S);
}

// ---------------------------------------------------------------------------
// Kernel 4: in-place streaming fixup of the V_TGT region:
// out[row, v] += corr[row], vectorized float4 (global_load/store_b128).
// ---------------------------------------------------------------------------
__global__ void fixup_kernel(float* __restrict__ out,
                             const float* __restrict__ corr) {
    int c4  = blockIdx.x * blockDim.x + threadIdx.x;   // float4 index in row
    int row = blockIdx.y;
    if (c4 >= VT / 4) return;
    float cr = corr[row];
    float4* p = (float4*)(out + (size_t)row * VOUT) + c4;
    float4 v = *p;
    v.x += cr; v.y += cr; v.z += cr; v.w += cr;
    *p = v;
}

// ---------------------------------------------------------------------------
// Kernel 5: ext-vocab scatter.  Rows in LDS (2000 f32 per row, 8 rows per
// block = 64000 B).  Serial per-row loop over SLEN keeps float accumulation
// order fixed -> bit-deterministic (atomics would not be).
// ---------------------------------------------------------------------------
__global__ void ext_kernel(const float* __restrict__ attn,
                           const int* __restrict__ copy_to_ext,
                           const float* __restrict__ oneminus,
                           float* __restrict__ out) {
    __shared__ float ldsacc[8 * VE];
    const int wv   = threadIdx.x >> 5;
    const int lane = threadIdx.x & 31;
    const int n    = blockIdx.x * 8 + wv;      // row (= t*BSZ + b)
    const int b    = n & (BSZ - 1);

    for (int i = threadIdx.x; i < 8 * VE; i += blockDim.x) ldsacc[i] = 0.f;
    __syncthreads();

    if (lane == 0) {
        float om = oneminus[n];
        const float* ar = attn + (size_t)n * SLEN;
        for (int s = 0; s < SLEN; ++s) {              // fixed order: deterministic
            int e = copy_to_ext[s * BSZ + b];
            if (e != 0) ldsacc[wv * VE + e] += ar[s] * om;   // UNK==0 ignored
        }
    }
    __syncthreads();

    size_t base = (size_t)n * VOUT + VT;
    for (int e = lane; e < VE; e += 32) {
        float v = fminf(fmaxf(ldsacc[wv * VE + e], 0.001f), 0.999f);
        out[base + e] = __logf(v);
    }
}

// ---------------------------------------------------------------------------
// Launch
// ---------------------------------------------------------------------------
extern "C" void kernel_launch(void* const* d_in, const int* in_sizes, int n_in,
                              void* d_out, int out_size, void* d_ws, size_t ws_size,
                              hipStream_t stream) {
    const float* hidden      = (const float*)d_in[0];   // [64,32,1024]
    const float* attn        = (const float*)d_in[1];   // [64,32,200]
    const int*   copy_to_ext = (const int*)  d_in[2];   // [200,32]
    const float* W_out       = (const float*)d_in[3];   // [32000,1024]
    const float* b_out       = (const float*)d_in[4];   // [32000]
    const float* w_copy      = (const float*)d_in[5];   // [1,1024]
    const float* b_copy      = (const float*)d_in[6];   // [1]
    float* out = (float*)d_out;                          // [2048, 34000]

    float* ws       = (float*)d_ws;
    float* copyv    = ws;                    // 2048
    float* logcopy  = ws + NROW;             // 2048
    float* oneminus = ws + 2 * NROW;         // 2048
    float* corr     = ws + 3 * NROW;         // 2048
    float* pM       = ws + 4 * NROW;         // 2048*400
    float* pS       = pM + (size_t)NROW * NPART;  // 2048*400  (~6.6 MB total)

    // 1) copy gate (one wave per row)
    gate_kernel<<<(NROW * 32) / 256, 256, 0, stream>>>(
        hidden, w_copy, b_copy, copyv, logcopy, oneminus);

    // 2) single GEMM pass: raw logits -> d_out, softmax partials -> ws
    dim3 gemm_grid(200, 8);
    gemm_kernel<<<gemm_grid, 256, 0, stream>>>(
        hidden, W_out, b_out, out, pM, pS);

    // 3) fold partials into per-row correction
    stats_reduce_kernel<<<(NROW + 255) / 256, 256, 0, stream>>>(
        pM, pS, logcopy, corr);

    // 4) streaming in-place fixup of the 262 MB logits region (b128 traffic)
    dim3 fix_grid((VT / 4 + 255) / 256, NROW);
    fixup_kernel<<<fix_grid, 256, 0, stream>>>(out, corr);

    // 5) ext-vocab scatter + safe_log  (last 2000 cols of every row)
    ext_kernel<<<NROW / 8, 256, 0, stream>>>(attn, copy_to_ext, oneminus, out);
}